// LPL_10316511445179
// MI455X (gfx1250) — compile-verified
//
#include <hip/hip_runtime.h>
#include <math.h>

// Problem sizes (fixed by the reference)
#define NB 256      // B
#define NC 1000     // C
#define ND 128      // D
#define NQ 262144   // Q

// Output layout (flat float offsets, reference return order)
#define O_OUT   0
#define O_OUT2  (NB*NC)
#define O_FEAT  (2*NB*NC)
#define O_PL    (O_FEAT + (2*NB + NQ)*ND)
#define O_SCORE (O_PL + (2*NB + NQ))
#define O_PROT  (O_SCORE + NB*NC)

typedef __attribute__((ext_vector_type(2))) float v2f;
typedef __attribute__((ext_vector_type(8))) float v8f;

// ---------------------------------------------------------------------------
// 1) pseudo_labels_b[i] = argmax_c output[i,c]*partial_Y[i,c]  (first index on tie)
//    One wave (32 lanes) per row. Also writes the two float label segments.
// ---------------------------------------------------------------------------
__global__ void argmax_labels_kernel(const float* __restrict__ out,
                                     const float* __restrict__ mask,
                                     int* __restrict__ labels,
                                     float* __restrict__ pl) {
    const int row  = blockIdx.x;
    const int lane = threadIdx.x;            // 0..31
    const float* o = out  + row * NC;
    const float* m = mask + row * NC;

    float bv = -INFINITY;
    int   bi = 0x7fffffff;
    for (int c = lane; c < NC; c += 32) {
        float v = o[c] * m[c];
        if (v > bv) { bv = v; bi = c; }      // strict > keeps first index
    }
    #pragma unroll
    for (int off = 16; off > 0; off >>= 1) {
        float ov = __shfl_down(bv, off, 32);
        int   oi = __shfl_down(bi, off, 32);
        if (ov > bv || (ov == bv && oi < bi)) { bv = ov; bi = oi; }
    }
    if (lane == 0) {
        labels[row]   = bi;
        pl[row]       = (float)bi;           // first B entries
        pl[NB + row]  = (float)bi;           // second B entries
    }
}

// ---------------------------------------------------------------------------
// 2) logits = q @ prototypes^T via V_WMMA_F32_16X16X4_F32.
//    One wave per 16x16 output tile; K-loop of 32 WMMAs (K=4 each).
//    grid = (M/16, ceil(N/16)) = (16, 63), block = 32 (full wave, EXEC all-1s).
// ---------------------------------------------------------------------------
__global__ void gemm_wmma_f32_kernel(const float* __restrict__ q,
                                     const float* __restrict__ protos,
                                     float* __restrict__ logits) {
    const int m0   = blockIdx.x * 16;
    const int n0   = blockIdx.y * 16;
    const int lane = threadIdx.x;            // 0..31
    const int r16  = lane & 15;
    const int kh   = lane >> 4;              // 0: K={0,1}, 1: K={2,3} per step

    // A: row (m0+r16) of q, K offset 2*kh
    const float* aptr = q + (m0 + r16) * ND + 2 * kh;
    // B: row (n0+r16) of prototypes (B matrix is K x N; col n = r16),
    // clamp row for the partial last N tile (stores are guarded below).
    int nrow = n0 + r16;
    if (nrow > NC - 1) nrow = NC - 1;
    const float* bptr = protos + nrow * ND + 2 * kh;

    v8f acc = {};
    #pragma unroll 8
    for (int k = 0; k < ND; k += 4) {
        v2f a, b;
        a.x = aptr[k];
        a.y = aptr[k + 1];
        b.x = bptr[k];
        b.y = bptr[k + 1];
        // D = A x B + C ; emits v_wmma_f32_16x16x4_f32
        acc = __builtin_amdgcn_wmma_f32_16x16x4_f32(
            /*neg_a=*/false, a, /*neg_b=*/false, b,
            /*c_mod=*/(short)0, acc, /*reuse_a=*/false, /*reuse_b=*/false);
    }

    // C/D layout: VGPR r -> M = r (lanes 0-15) or 8+r (lanes 16-31), N = lane&15
    const int n = n0 + r16;
    #pragma unroll
    for (int r = 0; r < 8; ++r) {
        int m = (lane < 16) ? r : (8 + r);
        if (n < NC) logits[(m0 + m) * NC + n] = acc[r];
    }
}

// ---------------------------------------------------------------------------
// 3) In-place row softmax over NC=1000 columns. One 256-thread block per row.
// ---------------------------------------------------------------------------
__global__ void softmax_rows_kernel(float* __restrict__ data) {
    __shared__ float red[256];
    const int row = blockIdx.x;
    const int tid = threadIdx.x;
    float* p = data + row * NC;

    float lmax = -INFINITY;
    for (int c = tid; c < NC; c += 256) lmax = fmaxf(lmax, p[c]);
    red[tid] = lmax; __syncthreads();
    for (int s = 128; s > 0; s >>= 1) {
        if (tid < s) red[tid] = fmaxf(red[tid], red[tid + s]);
        __syncthreads();
    }
    const float mx = red[0];
    __syncthreads();

    float lsum = 0.0f;
    for (int c = tid; c < NC; c += 256) {
        float e = expf(p[c] - mx);
        p[c] = e;
        lsum += e;
    }
    red[tid] = lsum; __syncthreads();
    for (int s = 128; s > 0; s >>= 1) {
        if (tid < s) red[tid] += red[tid + s];
        __syncthreads();
    }
    const float inv = 1.0f / red[0];
    for (int c = tid; c < NC; c += 256) p[c] *= inv;
}

// ---------------------------------------------------------------------------
// 4) Sequential EMA scatter (order-dependent) + L2 row normalize.
//    One 128-thread block per class; scan the 256 samples in order.
// ---------------------------------------------------------------------------
__global__ void ema_normalize_kernel(const float* __restrict__ q,
                                     const float* __restrict__ protos_in,
                                     const int* __restrict__ labels,
                                     float* __restrict__ protos_out) {
    const int c = blockIdx.x;
    const int d = threadIdx.x;               // 0..127
    const float m  = 0.99f;
    const float om = (float)(1.0 - 0.99);    // match reference's (1 - proto_m)

    float v = protos_in[c * ND + d];
    for (int i = 0; i < NB; ++i) {
        if (labels[i] == c)                  // uniform across block: no divergence
            v = v * m + om * q[i * ND + d];
    }

    __shared__ float red[128];
    red[d] = v * v; __syncthreads();
    for (int s = 64; s > 0; s >>= 1) {
        if (d < s) red[d] += red[d + s];
        __syncthreads();
    }
    float norm = sqrtf(red[0]);
    norm = fmaxf(norm, 1e-12f);
    protos_out[c * ND + d] = v / norm;
}

// ---------------------------------------------------------------------------
extern "C" void kernel_launch(void* const* d_in, const int* in_sizes, int n_in,
                              void* d_out, int out_size, void* d_ws, size_t ws_size,
                              hipStream_t stream) {
    const float* q        = (const float*)d_in[0];
    const float* k        = (const float*)d_in[1];
    const float* output   = (const float*)d_in[2];
    const float* output2  = (const float*)d_in[3];
    const float* partialY = (const float*)d_in[4];
    const float* protos   = (const float*)d_in[5];
    const float* queue    = (const float*)d_in[6];
    const float* queue_ps = (const float*)d_in[7];

    float* out = (float*)d_out;
    int* labels = (int*)d_ws;                // 256 ints of scratch

    // --- bulk passthroughs: D2D memcpy nodes (bandwidth-optimal, capturable) ---
    hipMemcpyAsync(out + O_OUT,  output,  (size_t)NB * NC * sizeof(float),
                   hipMemcpyDeviceToDevice, stream);
    hipMemcpyAsync(out + O_OUT2, output2, (size_t)NB * NC * sizeof(float),
                   hipMemcpyDeviceToDevice, stream);
    hipMemcpyAsync(out + O_FEAT,                     q,     (size_t)NB * ND * sizeof(float),
                   hipMemcpyDeviceToDevice, stream);
    hipMemcpyAsync(out + O_FEAT + NB * ND,           k,     (size_t)NB * ND * sizeof(float),
                   hipMemcpyDeviceToDevice, stream);
    hipMemcpyAsync(out + O_FEAT + 2 * NB * ND,       queue, (size_t)NQ * ND * sizeof(float),
                   hipMemcpyDeviceToDevice, stream);
    hipMemcpyAsync(out + O_PL + 2 * NB,              queue_ps, (size_t)NQ * sizeof(float),
                   hipMemcpyDeviceToDevice, stream);

    // --- 1) pseudo labels (argmax of masked logits) ---
    argmax_labels_kernel<<<NB, 32, 0, stream>>>(output, partialY, labels, out + O_PL);

    // --- 2) prototype logits via f32 WMMA ---
    dim3 ggrid(NB / 16, (NC + 15) / 16);     // (16, 63)
    gemm_wmma_f32_kernel<<<ggrid, 32, 0, stream>>>(q, protos, out + O_SCORE);

    // --- 3) row softmax -> score_prot ---
    softmax_rows_kernel<<<NB, 256, 0, stream>>>(out + O_SCORE);

    // --- 4) EMA scatter + L2 normalize -> protos ---
    ema_normalize_kernel<<<NC, ND, 0, stream>>>(q, protos, labels, out + O_PROT);
}